// MessagePassing_34050500723457
// MI455X (gfx1250) — compile-verified
//
#include <hip/hip_runtime.h>
#include <hip/hip_bf16.h>
#include <math.h>

// Problem constants (from reference)
#define NN     5000
#define EE     40000
#define UU     64
#define EFF    16
#define NSTEPS 4

typedef float v2f __attribute__((ext_vector_type(2)));
typedef float v4f __attribute__((ext_vector_type(4)));
typedef float v8f __attribute__((ext_vector_type(8)));

__device__ __forceinline__ v8f wmma_f32_16x16x4(v2f a, v2f b, v8f c) {
    // D = A(16x4) * B(4x16) + C(16x16), full fp32 via CDNA5 WMMA.
    return __builtin_amdgcn_wmma_f32_16x16x4_f32(
        false, a, false, b, (short)0, c, false, false);
}

// ---------------------------------------------------------------------------
// Kernel 1: edge_mats[e, i, j] = sum_f ef[e,f] * EK[f, i*64+j] + bias[i*64+j]
// GEMM (40000x16) @ (16x4096). One wave per 16x16 output tile, K=16 -> 4 WMMAs.
// em is write-once / streamed later; stores are non-temporal to spare L2.
// ---------------------------------------------------------------------------
__global__ __launch_bounds__(256) void edge_mats_kernel(
    const float* __restrict__ ef,     // E x 16
    const float* __restrict__ ek,     // 16 x 4096
    const float* __restrict__ ebias,  // 4096
    float* __restrict__ em)           // E x 4096
{
    const int lane = threadIdx.x & 31;
    const int wave = blockIdx.x * (blockDim.x >> 5) + (threadIdx.x >> 5);
    const int NT_N = (UU * UU) / 16;      // 256 col tiles
    const int NT_E = EE / 16;             // 2500 row tiles
    if (wave >= NT_E * NT_N) return;      // wave-uniform guard
    const int et = wave / NT_N;
    const int nt = wave % NT_N;
    const int e0 = et * 16, n0 = nt * 16;
    const int row = lane & 15;            // M for A, N for B/C
    const int hi  = lane >> 4;

    v8f c = {};
#pragma unroll
    for (int k0 = 0; k0 < EFF; k0 += 4) {
        const int ka = k0 + 2 * hi;       // K index for this lane half
        v2f a, b;
        a.x = ef[(e0 + row) * EFF + ka];
        a.y = ef[(e0 + row) * EFF + ka + 1];
        b.x = ek[(size_t)ka * (UU * UU) + n0 + row];
        b.y = ek[(size_t)(ka + 1) * (UU * UU) + n0 + row];
        c = wmma_f32_16x16x4(a, b, c);
    }
    const float bia = ebias[n0 + row];
#pragma unroll
    for (int r = 0; r < 8; ++r) {
        const int er = e0 + r + 8 * hi;   // C layout: VGPR r -> M=r / M=r+8
        __builtin_nontemporal_store(c[r] + bia,
                                    &em[(size_t)er * (UU * UU) + n0 + row]);
    }
}

// ---------------------------------------------------------------------------
// Kernel 2: zero the aggregation buffer
// ---------------------------------------------------------------------------
__global__ void zero_kernel(float* __restrict__ p, int n) {
    const int i = blockIdx.x * blockDim.x + threadIdx.x;
    if (i < n) p[i] = 0.0f;
}

// ---------------------------------------------------------------------------
// Kernel 3: per-edge matvec + scatter. HBM-bound (streams 655 MB of em).
// 256 threads = 4 edges x 64 threads. Neighbor vector broadcast via LDS.
// em loads are non-temporal (no reuse within or across steps > L2 capacity);
// h / agg / pairs keep regular-temporal so their reuse stays L2-resident.
// ---------------------------------------------------------------------------
__global__ __launch_bounds__(256) void msg_kernel(
    const float* __restrict__ em,      // E x 64 x 64
    const float* __restrict__ h,       // N x 64
    const int*   __restrict__ pairs,   // E x 2 (src, nbr)
    float* __restrict__ agg)           // N x 64
{
    __shared__ float nb[4][UU];
    const int sub = threadIdx.x >> 6;        // edge sub-group 0..3
    const int g   = threadIdx.x & 63;        // output row 0..63
    const int e   = blockIdx.x * 4 + sub;    // E % 4 == 0, no guard needed

    const int src = pairs[e * 2 + 0];
    const int nbr = pairs[e * 2 + 1];
    nb[sub][g] = h[(size_t)nbr * UU + g];
    __syncthreads();

    const v4f* mrow =
        reinterpret_cast<const v4f*>(em + (size_t)e * (UU * UU) + (size_t)g * UU);
    float acc = 0.0f;
#pragma unroll
    for (int j = 0; j < UU / 4; ++j) {
        const v4f m = __builtin_nontemporal_load(&mrow[j]);
        acc += m.x * nb[sub][4 * j + 0] + m.y * nb[sub][4 * j + 1] +
               m.z * nb[sub][4 * j + 2] + m.w * nb[sub][4 * j + 3];
    }
    atomicAdd(&agg[(size_t)src * UU + g], acc);
}

// ---------------------------------------------------------------------------
// Kernel 4: fused GRU cell via WMMA.
// One wave per 16 nodes. A-fragments (agg rows, h rows) loaded once; 6 WMMA
// accumulation chains (mx/mh at z,r,h column blocks) per 16-col tile, K=64.
// Gate math is done lane-wise directly on the C-layout, then stored.
// ---------------------------------------------------------------------------
__global__ __launch_bounds__(256) void gru_kernel(
    const float* __restrict__ agg,   // N x 64   (x input)
    const float* __restrict__ hin,   // N x 64   (recurrent input)
    const float* __restrict__ gk,    // 64 x 192
    const float* __restrict__ grk,   // 64 x 192
    const float* __restrict__ gb,    // 2 x 192
    float* __restrict__ hout,        // N x 64
    int n)
{
    const int lane = threadIdx.x & 31;
    const int wave = blockIdx.x * (blockDim.x >> 5) + (threadIdx.x >> 5);
    const int ntiles = (n + 15) / 16;
    if (wave >= ntiles) return;               // wave-uniform guard
    const int m0  = wave * 16;
    const int row = lane & 15;
    const int hi  = lane >> 4;

    // A fragments: lane holds row (m0+row), K = 4k+2*hi .. +1
    const int node = min(m0 + row, n - 1);    // clamp for partial last tile
    v2f aX[16], aH[16];
#pragma unroll
    for (int k = 0; k < 16; ++k) {
        const int ka = 4 * k + 2 * hi;
        aX[k].x = agg[(size_t)node * UU + ka];
        aX[k].y = agg[(size_t)node * UU + ka + 1];
        aH[k].x = hin[(size_t)node * UU + ka];
        aH[k].y = hin[(size_t)node * UU + ka + 1];
    }

#pragma unroll 1
    for (int t = 0; t < UU / 16; ++t) {       // 4 column tiles of h
        const int cz = 0 * UU + t * 16 + row; // z columns in [0,64)
        const int cr = 1 * UU + t * 16 + row; // r columns in [64,128)
        const int ch = 2 * UU + t * 16 + row; // h columns in [128,192)

        v8f xz = {}, xr = {}, xh = {}, iz = {}, ir = {}, ih = {};
#pragma unroll
        for (int k = 0; k < 16; ++k) {
            const int ka = 4 * k + 2 * hi;
            v2f b;
            b.x = gk[ka * 192 + cz];  b.y = gk[(ka + 1) * 192 + cz];
            xz = wmma_f32_16x16x4(aX[k], b, xz);
            b.x = gk[ka * 192 + cr];  b.y = gk[(ka + 1) * 192 + cr];
            xr = wmma_f32_16x16x4(aX[k], b, xr);
            b.x = gk[ka * 192 + ch];  b.y = gk[(ka + 1) * 192 + ch];
            xh = wmma_f32_16x16x4(aX[k], b, xh);
            b.x = grk[ka * 192 + cz]; b.y = grk[(ka + 1) * 192 + cz];
            iz = wmma_f32_16x16x4(aH[k], b, iz);
            b.x = grk[ka * 192 + cr]; b.y = grk[(ka + 1) * 192 + cr];
            ir = wmma_f32_16x16x4(aH[k], b, ir);
            b.x = grk[ka * 192 + ch]; b.y = grk[(ka + 1) * 192 + ch];
            ih = wmma_f32_16x16x4(aH[k], b, ih);
        }

        const float bxz = gb[cz],       bxr = gb[cr],       bxh = gb[ch];
        const float biz = gb[192 + cz], bir = gb[192 + cr], bih = gb[192 + ch];
        const int   col = t * 16 + row;           // column in h (0..63)

#pragma unroll
        for (int r = 0; r < 8; ++r) {
            const int nrow = m0 + r + 8 * hi;     // C layout: M = r / r+8
            const int nrc  = min(nrow, n - 1);
            const float hold = hin[(size_t)nrc * UU + col];
            const float z  = 1.0f / (1.0f + expf(-((xz[r] + bxz) + (iz[r] + biz))));
            const float rg = 1.0f / (1.0f + expf(-((xr[r] + bxr) + (ir[r] + bir))));
            const float hh = tanhf((xh[r] + bxh) + rg * (ih[r] + bih));
            const float hnew = z * hold + (1.0f - z) * hh;
            if (nrow < n) hout[(size_t)nrow * UU + col] = hnew;
        }
    }
}

// ---------------------------------------------------------------------------
// Host-side launch. Workspace layout (floats):
//   em  : E*U*U   = 163,840,000  (655.36 MB)  precomputed edge matrices
//   agg : N*U     = 320,000
//   hA  : N*U
//   hB  : N*U
// Total ~659 MB. Ping-pong h buffers; final step writes d_out directly.
// ---------------------------------------------------------------------------
extern "C" void kernel_launch(void* const* d_in, const int* in_sizes, int n_in,
                              void* d_out, int out_size, void* d_ws, size_t ws_size,
                              hipStream_t stream)
{
    const float* node_features = (const float*)d_in[0];
    const float* edge_features = (const float*)d_in[1];
    const int*   pair_indices  = (const int*)d_in[2];
    const float* edge_kernel   = (const float*)d_in[3];
    const float* edge_bias     = (const float*)d_in[4];
    const float* gru_k         = (const float*)d_in[5];
    const float* gru_rk        = (const float*)d_in[6];
    const float* gru_bias      = (const float*)d_in[7];

    float* em  = (float*)d_ws;
    float* agg = em  + (size_t)EE * UU * UU;
    float* hA  = agg + (size_t)NN * UU;
    float* hB  = hA  + (size_t)NN * UU;

    // 1) Precompute edge matrices (WMMA f32 GEMM, store-bound, NT stores).
    {
        const int waves  = (EE / 16) * ((UU * UU) / 16); // 640,000
        const int blocks = waves / 8;                    // 8 waves / 256-thr block
        edge_mats_kernel<<<blocks, 256, 0, stream>>>(edge_features, edge_kernel,
                                                     edge_bias, em);
    }

    const int gtiles  = (NN + 15) / 16;
    const int gblocks = (gtiles + 7) / 8;

    const float* hin = node_features;
    for (int s = 0; s < NSTEPS; ++s) {
        zero_kernel<<<(NN * UU + 255) / 256, 256, 0, stream>>>(agg, NN * UU);
        msg_kernel<<<EE / 4, 256, 0, stream>>>(em, hin, pair_indices, agg);
        float* hout = (s == NSTEPS - 1) ? (float*)d_out : ((s & 1) ? hB : hA);
        gru_kernel<<<gblocks, 256, 0, stream>>>(agg, hin, gru_k, gru_rk, gru_bias,
                                                hout, NN);
        hin = hout;
    }
}